// pooling_17721035063422
// MI455X (gfx1250) — compile-verified
//
#include <hip/hip_runtime.h>
#include <hip/hip_bf16.h>
#include <cstdint>

#define N_NODES 50000
#define N_EDGES 1600000
#define D_FEAT  128

typedef float v2f __attribute__((ext_vector_type(2)));
typedef float v8f __attribute__((ext_vector_type(8)));

// Exact float min/max via integer atomics (matches IEEE ordering, handles the
// +inf / -inf segment identities). Positive floats order like signed ints,
// negative floats order reversed as unsigned ints.
__device__ __forceinline__ void atomic_min_f32(float* addr, float v) {
    if (v >= 0.0f) atomicMin((int*)addr, __float_as_int(v));
    else           atomicMax((unsigned int*)addr, __float_as_uint(v));
}
__device__ __forceinline__ void atomic_max_f32(float* addr, float v) {
    if (v >= 0.0f) atomicMax((int*)addr, __float_as_int(v));
    else           atomicMin((unsigned int*)addr, __float_as_uint(v));
}

// ---------------------------------------------------------------------------
// Kernel 0: reset accumulators (run every launch; harness never re-poisons)
// and fold w_sum[d] = W[d] + W[d+128].
// ---------------------------------------------------------------------------
__global__ void init_kernel(const float* __restrict__ W,
                            float* __restrict__ node_min,
                            float* __restrict__ node_max,
                            float* __restrict__ node_sum,
                            unsigned int* __restrict__ node_deg,
                            float* __restrict__ wsum) {
    int i = blockIdx.x * blockDim.x + threadIdx.x;
    if (i < N_NODES) {
        node_min[i] = __int_as_float(0x7f800000);   // +inf (segment_min identity)
        node_max[i] = __int_as_float(0xff800000);   // -inf (segment_max identity)
        node_sum[i] = 0.0f;
        node_deg[i] = 0u;
    }
    if (i < D_FEAT) wsum[i] = W[i] + W[i + D_FEAT];
}

// ---------------------------------------------------------------------------
// Kernel 1: t[node] = dot(x[node], wsum) via V_WMMA_F32_16X16X4_F32.
// One wave handles a 16-row tile of x. A = 16x4 chunk of x, B = wsum chunk
// replicated across all 16 columns, so D[m][n] == t[base+m] for every n.
// ISA 32-bit A 16x4 layout: lanes 0-15 -> M=lane, VGPR0/1 = K0/K1;
// lanes 16-31 -> M=lane-16, VGPR0/1 = K2/K3. B is the lane=column analogue.
// ---------------------------------------------------------------------------
__global__ __launch_bounds__(256)
void node_proj_kernel(const float* __restrict__ x,
                      const float* __restrict__ wsum,
                      float* __restrict__ t) {
    const int lane = threadIdx.x & 31;
    const int tile = blockIdx.x * 8 + (threadIdx.x >> 5);
    const int ntiles = (N_NODES + 15) / 16;
    if (tile >= ntiles) return;                 // wave-uniform: EXEC stays all-1s

    const int base = tile * 16;
    const int half = lane >> 4;                 // 0: K pair {0,1}, 1: K pair {2,3}
    const float* xrow = x + (size_t)(base + (lane & 15)) * D_FEAT;

    // Pull this row's 4 cachelines (512B) toward the WGP before the MAC loop.
    __builtin_prefetch(xrow + half * 64, 0, 3);
    __builtin_prefetch(xrow + half * 64 + 32, 0, 3);

    v8f acc = {};
#pragma unroll 4
    for (int k = 0; k < D_FEAT; k += 4) {
        const int kk = k + half * 2;
        v2f a  = *(const v2f*)(xrow + kk);      // A: x tile chunk (16x4)
        v2f bm = *(const v2f*)(wsum + kk);      // B: w chunk, same in all 16 cols
        acc = __builtin_amdgcn_wmma_f32_16x16x4_f32(
                  false, a, false, bm, (short)0, acc, false, false);
    }

    // D layout: VGPR r -> M=r (lanes 0-15), M=r+8 (lanes 16-31); all N equal.
    const int m = lane & 7;
    float tv = acc[0];
#pragma unroll
    for (int r = 1; r < 8; ++r) tv = (m == r) ? acc[r] : tv;
    if ((lane & 8) == 0) {                      // lanes 0-7 -> rows 0-7, 16-23 -> rows 8-15
        int node = base + half * 8 + m;
        if (node < N_NODES) t[node] = tv;
    }
}

// ---------------------------------------------------------------------------
// Kernel 2: edge scores + scatter min/max/sum/count to BOTH endpoints.
// ---------------------------------------------------------------------------
__global__ __launch_bounds__(256)
void edge_kernel(const int* __restrict__ src, const int* __restrict__ dst,
                 const float* __restrict__ t, const float* __restrict__ b,
                 float* __restrict__ edge_score,
                 float* __restrict__ node_min, float* __restrict__ node_max,
                 float* __restrict__ node_sum, unsigned int* __restrict__ node_deg) {
    int e = blockIdx.x * blockDim.x + threadIdx.x;
    if (e >= N_EDGES) return;
    int s = src[e];
    int d = dst[e];
    float sc = t[s] + t[d] + b[0];
    edge_score[e] = sc;

    atomic_min_f32(node_min + s, sc);
    atomic_min_f32(node_min + d, sc);
    atomic_max_f32(node_max + s, sc);
    atomic_max_f32(node_max + d, sc);
    atomicAdd(node_sum + s, sc);
    atomicAdd(node_sum + d, sc);
    atomicAdd(node_deg + s, 1u);
    atomicAdd(node_deg + d, 1u);
}

// ---------------------------------------------------------------------------
// Kernel 3: avg = sum / max(deg, 1)
// ---------------------------------------------------------------------------
__global__ void finalize_kernel(const float* __restrict__ node_sum,
                                const unsigned int* __restrict__ node_deg,
                                float* __restrict__ node_avg) {
    int i = blockIdx.x * blockDim.x + threadIdx.x;
    if (i < N_NODES) {
        float dg = (float)node_deg[i];
        node_avg[i] = node_sum[i] / fmaxf(dg, 1.0f);
    }
}

extern "C" void kernel_launch(void* const* d_in, const int* in_sizes, int n_in,
                              void* d_out, int out_size, void* d_ws, size_t ws_size,
                              hipStream_t stream) {
    const int*   edge_index = (const int*)d_in[0];   // [2, E]
    const float* x          = (const float*)d_in[1]; // [N, 128]
    const float* W          = (const float*)d_in[2]; // [1, 256]
    const float* b          = (const float*)d_in[3]; // [1]
    const int* src = edge_index;
    const int* dst = edge_index + N_EDGES;

    // d_out = [edge_score(E), node_min(N), node_max(N), node_avg(N)]
    float* out        = (float*)d_out;
    float* edge_score = out;
    float* node_min   = out + N_EDGES;
    float* node_max   = node_min + N_NODES;
    float* node_avg   = node_max + N_NODES;

    // workspace: [t(N f32)][sum(N f32)][deg(N u32)][wsum(128 f32)]
    float*        t        = (float*)d_ws;
    float*        node_sum = t + N_NODES;
    unsigned int* node_deg = (unsigned int*)(node_sum + N_NODES);
    float*        wsum     = (float*)(node_deg + N_NODES);

    init_kernel<<<(N_NODES + 255) / 256, 256, 0, stream>>>(
        W, node_min, node_max, node_sum, node_deg, wsum);

    const int ntiles = (N_NODES + 15) / 16;          // 3125 tiles, 8 waves/block
    node_proj_kernel<<<(ntiles + 7) / 8, 256, 0, stream>>>(x, wsum, t);

    edge_kernel<<<(N_EDGES + 255) / 256, 256, 0, stream>>>(
        src, dst, t, b, edge_score, node_min, node_max, node_sum, node_deg);

    finalize_kernel<<<(N_NODES + 255) / 256, 256, 0, stream>>>(
        node_sum, node_deg, node_avg);
}